// MotionGCN_29695403884881
// MI455X (gfx1250) — compile-verified
//
#include <hip/hip_runtime.h>
#include <hip/hip_bf16.h>

// ---------------------------------------------------------------------------
// MotionGCN on MI455X (gfx1250, wave32).
// Dense 16/32-wide GEMMs use V_WMMA_F32_16X16X4_F32 (native fp32 WMMA path).
// Sparse aggregation: one thread per edge, global_atomic_add_f32 into an
// aggregation buffer pre-initialized with bias + self-loop term.
// ReLU is fused into the consumer GEMM's A-operand loads.
// ---------------------------------------------------------------------------

typedef float v2f __attribute__((ext_vector_type(2)));
typedef float v8f __attribute__((ext_vector_type(8)));

#define N_NODES 491520
#define F_IN    16

// ---------------- init / degree kernels ----------------

__global__ void k_init_dis(float* __restrict__ dis, int n) {
    int i = blockIdx.x * blockDim.x + threadIdx.x;
    if (i < n) dis[i] = 1.0f;   // self-loop contributes 1 to in-degree
}

__global__ void k_degree(const long long* __restrict__ col, float* __restrict__ dis, int E) {
    int e = blockIdx.x * blockDim.x + threadIdx.x;
    if (e < E) atomicAdd(dis + col[e], 1.0f);
}

__global__ void k_rsqrt(float* __restrict__ dis, int n) {
    int i = blockIdx.x * blockDim.x + threadIdx.x;
    if (i < n) dis[i] = rsqrtf(dis[i]);   // deg >= 1 always (self loops)
}

// ---------------- WMMA GEMM kernels (fp32, 16x16x4) ----------------
// One wave computes one 16-row tile. Block = 128 threads = 4 waves.
// A frag (16x4 f32): lane l<16 holds rows M=l, K = {0,1}; lanes 16..31 K={2,3}.
// B frag (4x16 f32): mirrored on K; N = lane&15.
// C/D (16x16 f32): VGPR r = row M=r (lanes0-15) / M=r+8 (lanes16-31), N=lane&15.

// Y[N,16] = X[N,16] @ W[16,16]
__global__ void k_gemm_16to16(const float* __restrict__ X, const float* __restrict__ W,
                              float* __restrict__ Y) {
    int lane = threadIdx.x & 31;
    int wave = threadIdx.x >> 5;
    int tile = blockIdx.x * 4 + wave;
    int m    = lane & 15;
    int half = lane >> 4;
    int row0 = tile * 16;

    const float* xr = X + (size_t)(row0 + m) * 16;
    v8f c = {};
#pragma unroll
    for (int k4 = 0; k4 < 4; ++k4) {
        int kb = k4 * 4 + half * 2;
        v2f a, b;
        a.x = xr[kb];            a.y = xr[kb + 1];
        b.x = W[kb * 16 + m];    b.y = W[(kb + 1) * 16 + m];
        c = __builtin_amdgcn_wmma_f32_16x16x4_f32(false, a, false, b, (short)0, c, false, false);
    }
    float* yr = Y + (size_t)row0 * 16;
#pragma unroll
    for (int r = 0; r < 8; ++r)
        yr[(size_t)(r + half * 8) * 16 + m] = c[r];
}

// Y[N,32] = relu(X[N,16]) @ W[16,32]
__global__ void k_gemm_16to32_relu(const float* __restrict__ X, const float* __restrict__ W,
                                   float* __restrict__ Y) {
    int lane = threadIdx.x & 31;
    int wave = threadIdx.x >> 5;
    int tile = blockIdx.x * 4 + wave;
    int m    = lane & 15;
    int half = lane >> 4;
    int row0 = tile * 16;

    const float* xr = X + (size_t)(row0 + m) * 16;
    v8f c0 = {}, c1 = {};
#pragma unroll
    for (int k4 = 0; k4 < 4; ++k4) {
        int kb = k4 * 4 + half * 2;
        v2f a, b0, b1;
        a.x = fmaxf(xr[kb], 0.0f);
        a.y = fmaxf(xr[kb + 1], 0.0f);
        b0.x = W[kb * 32 + m];        b0.y = W[(kb + 1) * 32 + m];
        b1.x = W[kb * 32 + 16 + m];   b1.y = W[(kb + 1) * 32 + 16 + m];
        c0 = __builtin_amdgcn_wmma_f32_16x16x4_f32(false, a, false, b0, (short)0, c0, false, false);
        c1 = __builtin_amdgcn_wmma_f32_16x16x4_f32(false, a, false, b1, (short)0, c1, false, false);
    }
    float* yr = Y + (size_t)row0 * 32;
#pragma unroll
    for (int r = 0; r < 8; ++r) {
        yr[(size_t)(r + half * 8) * 32 + m]      = c0[r];
        yr[(size_t)(r + half * 8) * 32 + 16 + m] = c1[r];
    }
}

// OUT[N,16] = relu(X[N,32]) @ W[32,16] + bout[16]
__global__ void k_gemm_32to16_relu_bias(const float* __restrict__ X, const float* __restrict__ W,
                                        const float* __restrict__ bout, float* __restrict__ OUT) {
    int lane = threadIdx.x & 31;
    int wave = threadIdx.x >> 5;
    int tile = blockIdx.x * 4 + wave;
    int m    = lane & 15;
    int half = lane >> 4;
    int row0 = tile * 16;

    const float* xr = X + (size_t)(row0 + m) * 32;
    v8f c = {};
#pragma unroll
    for (int k4 = 0; k4 < 8; ++k4) {          // K = 32
        int kb = k4 * 4 + half * 2;
        v2f a, b;
        a.x = fmaxf(xr[kb], 0.0f);
        a.y = fmaxf(xr[kb + 1], 0.0f);
        b.x = W[kb * 16 + m];    b.y = W[(kb + 1) * 16 + m];
        c = __builtin_amdgcn_wmma_f32_16x16x4_f32(false, a, false, b, (short)0, c, false, false);
    }
    float bb = bout[m];
    float* yr = OUT + (size_t)row0 * 16;
#pragma unroll
    for (int r = 0; r < 8; ++r)
        yr[(size_t)(r + half * 8) * 16 + m] = c[r] + bb;
}

// ---------------- aggregation (self-loop + bias fused into init) ----------------

// agg[i][f] = b[f] + dis[i]^2 * h[i][f]   (F = 16 or 32, flat over N*F threads)
__global__ void k_init_agg(const float* __restrict__ h, const float* __restrict__ dis,
                           const float* __restrict__ b, float* __restrict__ agg,
                           int n, int F) {
    int i = blockIdx.x * blockDim.x + threadIdx.x;
    if (i >= n * F) return;
    int node = i / F;
    int f    = i - node * F;
    float d  = dis[node];
    agg[i] = b[f] + d * d * h[i];
}

// one thread per edge, F=16: 4x float4 gather + 16 float atomics
__global__ void k_scatter16(const float* __restrict__ h, const long long* __restrict__ row,
                            const long long* __restrict__ col, const float* __restrict__ dis,
                            float* __restrict__ agg, int E) {
    int e = blockIdx.x * blockDim.x + threadIdx.x;
    if (e >= E) return;
    long long r = row[e], c = col[e];
    float nrm = dis[r] * dis[c];
    const float4* hr = (const float4*)(h + (size_t)r * 16);
    float* out = agg + (size_t)c * 16;
#pragma unroll
    for (int q = 0; q < 4; ++q) {
        float4 v = hr[q];
        atomicAdd(out + q * 4 + 0, v.x * nrm);
        atomicAdd(out + q * 4 + 1, v.y * nrm);
        atomicAdd(out + q * 4 + 2, v.z * nrm);
        atomicAdd(out + q * 4 + 3, v.w * nrm);
    }
}

// one thread per edge, F=32
__global__ void k_scatter32(const float* __restrict__ h, const long long* __restrict__ row,
                            const long long* __restrict__ col, const float* __restrict__ dis,
                            float* __restrict__ agg, int E) {
    int e = blockIdx.x * blockDim.x + threadIdx.x;
    if (e >= E) return;
    long long r = row[e], c = col[e];
    float nrm = dis[r] * dis[c];
    const float4* hr = (const float4*)(h + (size_t)r * 32);
    float* out = agg + (size_t)c * 32;
#pragma unroll
    for (int q = 0; q < 8; ++q) {
        float4 v = hr[q];
        atomicAdd(out + q * 4 + 0, v.x * nrm);
        atomicAdd(out + q * 4 + 1, v.y * nrm);
        atomicAdd(out + q * 4 + 2, v.z * nrm);
        atomicAdd(out + q * 4 + 3, v.w * nrm);
    }
}

// ---------------------------------------------------------------------------

extern "C" void kernel_launch(void* const* d_in, const int* in_sizes, int n_in,
                              void* d_out, int out_size, void* d_ws, size_t ws_size,
                              hipStream_t stream) {
    const float*      x    = (const float*)d_in[0];
    const long long*  ei   = (const long long*)d_in[1];
    const float*      W1   = (const float*)d_in[2];
    const float*      b1   = (const float*)d_in[3];
    const float*      W2   = (const float*)d_in[4];
    const float*      b2   = (const float*)d_in[5];
    const float*      Wout = (const float*)d_in[6];
    const float*      bout = (const float*)d_in[7];
    float*            out  = (float*)d_out;

    const int n = in_sizes[0] / F_IN;        // 491520
    const int E = in_sizes[1] / 2;           // 2,000,000
    const long long* erow = ei;
    const long long* ecol = ei + E;

    // workspace layout (floats): dis | h1(N*16) | agg1(N*16) | h2(N*32) | agg2(N*32)
    float* dis  = (float*)d_ws;
    float* h1   = dis  + (size_t)n;
    float* agg1 = h1   + (size_t)n * 16;
    float* h2   = agg1 + (size_t)n * 16;
    float* agg2 = h2   + (size_t)n * 32;

    const int TB = 256;
    const int tiles  = n / 16;               // 30720 (exact)
    const int gblk   = tiles / 4;            // 4 waves per 128-thread block
    const int eblk   = (E + TB - 1) / TB;

    // degree -> dis
    k_init_dis<<<(n + TB - 1) / TB, TB, 0, stream>>>(dis, n);
    k_degree  <<<eblk, TB, 0, stream>>>(ecol, dis, E);
    k_rsqrt   <<<(n + TB - 1) / TB, TB, 0, stream>>>(dis, n);

    // layer 1: h1 = x @ W1 ; agg1 = b1 + selfloop + scatter
    k_gemm_16to16<<<gblk, 128, 0, stream>>>(x, W1, h1);
    k_init_agg<<<((n * 16) + TB - 1) / TB, TB, 0, stream>>>(h1, dis, b1, agg1, n, 16);
    k_scatter16<<<eblk, TB, 0, stream>>>(h1, erow, ecol, dis, agg1, E);

    // layer 2: h2 = relu(agg1) @ W2 ; agg2 = b2 + selfloop + scatter
    k_gemm_16to32_relu<<<gblk, 128, 0, stream>>>(agg1, W2, h2);
    k_init_agg<<<((n * 32) + TB - 1) / TB, TB, 0, stream>>>(h2, dis, b2, agg2, n, 32);
    k_scatter32<<<eblk, TB, 0, stream>>>(h2, erow, ecol, dis, agg2, E);

    // output: out = relu(agg2) @ Wout + bout
    k_gemm_32to16_relu_bias<<<gblk, 128, 0, stream>>>(agg2, Wout, bout, out);
}